// dnc_v2_5488968204597
// MI455X (gfx1250) — compile-verified
//
#include <hip/hip_runtime.h>
#include <hip/hip_bf16.h>
#include <math.h>

// ---------------- problem constants ----------------
#define Ndim   512
#define Wd     64
#define Rr     4
#define OUTD   256
#define IND    256
#define IFACE  471          // R*Wd + 3*Wd + 5*R + 3
#define CDIM   727          // OUT + IFACE
#define BSz    64
#define G3     2181         // 3*CDIM

// interface partition offsets (cumsum of [256,4,64,1,64,64,4,1,1,12])
#define OFF_KREAD   0
#define OFF_BREAD   256
#define OFF_KWRITE  260
#define OFF_BWRITE  324
#define OFF_ERASE   325
#define OFF_WRITEV  389
#define OFF_FREEG   453
#define OFF_ALLOCG  457
#define OFF_WRITEG  458
#define OFF_MODES   459

typedef float v2f  __attribute__((ext_vector_type(2)));
typedef float v4f  __attribute__((ext_vector_type(4)));
typedef float v8f  __attribute__((ext_vector_type(8)));
typedef unsigned int u32x4 __attribute__((ext_vector_type(4)));
typedef int   i32x8 __attribute__((ext_vector_type(8)));
typedef int   i32x4 __attribute__((ext_vector_type(4)));

__device__ __forceinline__ float sigmoidf_(float x) { return 1.0f / (1.0f + expf(-x)); }
__device__ __forceinline__ float softplusf_(float x) { return (x > 20.0f) ? x : log1pf(expf(x)); }
__device__ __forceinline__ float gelu_exact_(float x) {
    return 0.5f * x * (1.0f + erff(x * 0.70710678118654752f));
}

// ---------------------------------------------------------------------------
// Generic f32 GEMM via V_WMMA_F32_16X16X4_F32. One wave per 16x16 C tile.
// C[Mr x Nc] = A[Mr x K](row stride lda) @ B[K x Nc](row stride ldb) (+ bias)
// A-matrix lane layout (ISA 7.12.2): lane l -> m=l&15, VGPR0=A[m][(l>>4)*2],
// VGPR1=A[m][(l>>4)*2+1]. B assumed symmetric (lane -> n, same K split).
// C/D: VGPR v -> rows v (lanes 0-15) and v+8 (lanes 16-31).
// ---------------------------------------------------------------------------
__global__ __launch_bounds__(32) void k_gemm_wmma(
    const float* __restrict__ A, int lda,
    const float* __restrict__ B, int ldb,
    const float* __restrict__ bias,
    float* __restrict__ C, int ldc,
    int Mr, int Nc, int K)
{
    const int lane = threadIdx.x;
    const int nt = blockIdx.x * 16;
    const int mt = blockIdx.y * 16;
    const int nl = lane & 15;
    const int kh = (lane >> 4) * 2;

    const int m  = mt + nl;
    const int n  = nt + nl;
    const int mc = (m < Mr) ? m : (Mr - 1);   // clamped (safe) addresses
    const int nc = (n < Nc) ? n : (Nc - 1);
    const float am = (m < Mr) ? 1.0f : 0.0f;
    const float bm = (n < Nc) ? 1.0f : 0.0f;

    v8f acc = {};
    for (int k = 0; k < K; k += 4) {
        int k0 = k + kh, k1 = k0 + 1;
        int k0c = (k0 < K) ? k0 : (K - 1);
        int k1c = (k1 < K) ? k1 : (K - 1);
        float f0 = (k0 < K) ? 1.0f : 0.0f;
        float f1 = (k1 < K) ? 1.0f : 0.0f;
        v2f a, b;
        a.x = A[(size_t)mc * lda + k0c] * (am * f0);
        a.y = A[(size_t)mc * lda + k1c] * (am * f1);
        b.x = B[(size_t)k0c * ldb + nc] * (bm * f0);
        b.y = B[(size_t)k1c * ldb + nc] * (bm * f1);
        acc = __builtin_amdgcn_wmma_f32_16x16x4_f32(false, a, false, b,
                                                    (short)0, acc, false, false);
    }

    if (n < Nc) {
        float bv = bias ? bias[n] : 0.0f;
        int mbase = mt + (lane >> 4) * 8;
#pragma unroll
        for (int v = 0; v < 8; ++v) {
            int mm = mbase + v;
            if (mm < Mr) C[(size_t)mm * ldc + n] = acc[v] + bv;
        }
    }
}

// ---------------------------------------------------------------------------
// GRU gate combine: c = z*h + (1-z)*gelu(xh + r*rh)
// ---------------------------------------------------------------------------
__global__ void k_gru_combine(const float* __restrict__ mx,
                              const float* __restrict__ mh,
                              const float* __restrict__ hctrl,
                              float* __restrict__ c,
                              float* __restrict__ hc_out,
                              int head)
{
    int idx = blockIdx.x * blockDim.x + threadIdx.x;
    if (idx >= BSz * CDIM) return;
    int b = idx / CDIM, j = idx % CDIM;
    const float* mxr = mx + (size_t)b * G3;
    const float* mhr = mh + (size_t)b * G3;
    float z  = sigmoidf_(mxr[j]          + mhr[j]);
    float r  = sigmoidf_(mxr[CDIM + j]   + mhr[CDIM + j]);
    float hh = gelu_exact_(mxr[2*CDIM+j] + r * mhr[2*CDIM + j]);
    float hv = hctrl[(size_t)b * 2 * CDIM + head * CDIM + j];
    float cv = z * hv + (1.0f - z) * hh;
    c[(size_t)b * CDIM + j] = cv;
    hc_out[(size_t)b * 2 * CDIM + head * CDIM + j] = cv;
}

// ---------------------------------------------------------------------------
// Write head: usage update, bitonic-sort allocator, content lookup (old M),
// write weighting, M_n update, precedence update. One block per batch.
// ---------------------------------------------------------------------------
__global__ __launch_bounds__(512) void k_write_head(
    const float* __restrict__ if2, const float* __restrict__ M,
    const float* __restrict__ usage, const float* __restrict__ Wread,
    const float* __restrict__ Wwrite, const float* __restrict__ Wp,
    float* __restrict__ un_out, float* __restrict__ Ww_out,
    float* __restrict__ Wp_out, float* __restrict__ Mn_out,
    float* __restrict__ Ww_ws)
{
    __shared__ float sv[Ndim];   __shared__ int   si[Ndim];
    __shared__ float cp[Ndim];   __shared__ float walloc[Ndim];
    __shared__ float wwArr[Ndim];__shared__ float red[Ndim];
    __shared__ float eraseS[Wd], wvS[Wd], nkS[Wd];
    __shared__ float scal[8];

    const int b = blockIdx.x, tid = threadIdx.x;
    const float* f = if2 + (size_t)b * IFACE;

    if (tid < Wd) { eraseS[tid] = sigmoidf_(f[OFF_ERASE + tid]); wvS[tid] = f[OFF_WRITEV + tid]; }
    if (tid == 0) {
        scal[0] = 1.0f + softplusf_(f[OFF_BWRITE]);
        scal[1] = sigmoidf_(f[OFF_ALLOCG]);
        scal[2] = sigmoidf_(f[OFF_WRITEG]);
        for (int r = 0; r < Rr; ++r) scal[3 + r] = sigmoidf_(f[OFF_FREEG + r]);
    }
    __syncthreads();

    // retention + usage_n
    float ret = 1.0f;
#pragma unroll
    for (int r = 0; r < Rr; ++r)
        ret *= 1.0f - scal[3 + r] * Wread[((size_t)b * Ndim + tid) * Rr + r];
    float u  = usage[(size_t)b * Ndim + tid];
    float wv = Wwrite[(size_t)b * Ndim + tid];
    float un = (u + wv - u * wv) * ret;
    un_out[(size_t)b * Ndim + tid] = un;
    sv[tid] = un; si[tid] = tid;

    // bitonic sort ascending by (value, index)  == top_k(-usage) order
    for (int k2 = 2; k2 <= Ndim; k2 <<= 1)
        for (int j = k2 >> 1; j > 0; j >>= 1) {
            __syncthreads();
            int ixj = tid ^ j;
            if (ixj > tid) {
                float va = sv[tid], vb = sv[ixj];
                int   ia = si[tid], ib = si[ixj];
                bool aGTb = (va > vb) || (va == vb && ia > ib);
                bool up = ((tid & k2) == 0);
                if (up ? aGTb : !aGTb) { sv[tid]=vb; si[tid]=ib; sv[ixj]=va; si[ixj]=ia; }
            }
        }
    __syncthreads();

    // exclusive cumulative product of sorted usage
    cp[tid] = sv[tid]; __syncthreads();
    for (int off = 1; off < Ndim; off <<= 1) {
        float t = (tid >= off) ? cp[tid - off] : 1.0f;
        __syncthreads();
        cp[tid] *= t;
        __syncthreads();
    }
    float excl = (tid == 0) ? 1.0f : cp[tid - 1];
    walloc[si[tid]] = (1.0f - sv[tid]) * excl;

    // write-key normalization
    if (tid < Wd) red[tid] = f[OFF_KWRITE + tid] * f[OFF_KWRITE + tid];
    __syncthreads();
    if (tid == 0) {
        float s = 0.0f;
        for (int w = 0; w < Wd; ++w) s += red[w];
        scal[7] = 1.0f / sqrtf(fmaxf(s, 1e-12f));
    }
    __syncthreads();
    if (tid < Wd) nkS[tid] = f[OFF_KWRITE + tid] * scal[7];
    __syncthreads();

    // cosine similarity against old M, softmax over N
    const float* mrow = M + ((size_t)b * Ndim + tid) * Wd;
    float dot = 0.0f, msq = 0.0f;
    for (int w = 0; w < Wd; ++w) { float mvv = mrow[w]; dot += mvv * nkS[w]; msq += mvv * mvv; }
    float sim = dot * (1.0f / sqrtf(fmaxf(msq, 1e-12f))) * scal[0];

    red[tid] = sim; __syncthreads();
    for (int s = Ndim/2; s > 0; s >>= 1) { if (tid < s) red[tid] = fmaxf(red[tid], red[tid+s]); __syncthreads(); }
    float mxv = red[0]; __syncthreads();
    float e = expf(sim - mxv);
    red[tid] = e; __syncthreads();
    for (int s = Ndim/2; s > 0; s >>= 1) { if (tid < s) red[tid] += red[tid+s]; __syncthreads(); }
    float wl = e / red[0]; __syncthreads();

    // write weighting
    float ww = scal[2] * (scal[1] * walloc[tid] + (1.0f - scal[1]) * wl);
    wwArr[tid] = ww;
    Ww_out[(size_t)b * Ndim + tid] = ww;
    Ww_ws [(size_t)b * Ndim + tid] = ww;

    red[tid] = ww; __syncthreads();
    for (int s = Ndim/2; s > 0; s >>= 1) { if (tid < s) red[tid] += red[tid+s]; __syncthreads(); }
    float swsum = red[0]; __syncthreads();
    Wp_out[(size_t)b * Ndim + tid] = (1.0f - swsum) * Wp[(size_t)b * Ndim + tid] + ww;

    // M_n = M*(1 - ww*erase) + ww*write_v
    for (int idx = tid; idx < Ndim * Wd; idx += 512) {
        int n = idx >> 6, w = idx & 63;
        float mvv = M[(size_t)b * Ndim * Wd + idx];
        Mn_out[(size_t)b * Ndim * Wd + idx] =
            mvv * (1.0f - wwArr[n] * eraseS[w]) + wwArr[n] * wvS[w];
    }
}

__global__ void k_zero(float* __restrict__ p, int n) {
    int i = blockIdx.x * blockDim.x + threadIdx.x;
    if (i < n) p[i] = 0.0f;
}

// ---------------------------------------------------------------------------
// Fused L_n streaming kernel (the HBM-dominant stage, ~128MB traffic):
// reads L once (NT), writes L_n (NT), and accumulates W_fwd = L_n @ W_read
// (row-local) and W_bwd = L_n^T @ W_read (LDS partials + global atomics).
// Block = (batch b, 32-row tile); 8 threads per row cover 512 cols (v4f).
// ---------------------------------------------------------------------------
__global__ __launch_bounds__(256) void k_Ln(
    const float* __restrict__ L, const float* __restrict__ Ww_ws,
    const float* __restrict__ Wp, const float* __restrict__ Wread,
    float* __restrict__ Ln_out, float* __restrict__ Wfwd, float* __restrict__ Wbwd)
{
    __shared__ float wwS[Ndim], wpS[Ndim];
    __shared__ float wrS[Ndim * Rr];
    __shared__ float bwdS[Ndim * Rr];
    __shared__ float redS[32][8][4];

    const int b = blockIdx.x, rt = blockIdx.y, tid = threadIdx.x;
    for (int i = tid; i < Ndim; i += 256) {
        wwS[i] = Ww_ws[(size_t)b * Ndim + i];
        wpS[i] = Wp  [(size_t)b * Ndim + i];
    }
    for (int i = tid; i < Ndim * Rr; i += 256) {
        wrS[i] = Wread[(size_t)b * Ndim * Rr + i];
        bwdS[i] = 0.0f;
    }
    __syncthreads();

    const int ri  = rt * 32 + (tid >> 3);
    const int sub = tid & 7;
    const float wwi = wwS[ri];
    float wri[4];
#pragma unroll
    for (int r = 0; r < 4; ++r) wri[r] = wrS[ri * Rr + r];
    float fwd[4] = {0.f, 0.f, 0.f, 0.f};

    const float* lrow = L      + ((size_t)b * Ndim + ri) * Ndim;
    float*       lout = Ln_out + ((size_t)b * Ndim + ri) * Ndim;

    const int j0 = sub * 64;
    for (int j = j0; j < j0 + 64; j += 4) {
        __builtin_prefetch(lrow + j + 32, 0, 0);
        v4f lv = __builtin_nontemporal_load((const v4f*)(lrow + j));
        float vv[4] = {lv.x, lv.y, lv.z, lv.w};
#pragma unroll
        for (int c = 0; c < 4; ++c) {
            int jj = j + c;
            float v = (1.0f - wwi - wwS[jj]) * vv[c] + wwi * wpS[jj];
            if (jj == ri) v = 0.0f;
            vv[c] = v;
#pragma unroll
            for (int r = 0; r < 4; ++r) {
                fwd[r] += v * wrS[jj * Rr + r];
                atomicAdd(&bwdS[jj * Rr + r], v * wri[r]);
            }
        }
        v4f ov = {vv[0], vv[1], vv[2], vv[3]};
        __builtin_nontemporal_store(ov, (v4f*)(lout + j));
    }

    // reduce fwd across 8 col-threads of each row
#pragma unroll
    for (int r = 0; r < 4; ++r) redS[tid >> 3][sub][r] = fwd[r];
    __syncthreads();
    if (sub == 0) {
#pragma unroll
        for (int r = 0; r < 4; ++r) {
            float s = 0.0f;
            for (int k = 0; k < 8; ++k) s += redS[tid >> 3][k][r];
            Wfwd[((size_t)b * Ndim + ri) * Rr + r] = s;
        }
    }
    __syncthreads();
    for (int i = tid; i < Ndim * Rr; i += 256)
        atomicAdd(&Wbwd[(size_t)b * Ndim * Rr + i], bwdS[i]);
}

// ---------------------------------------------------------------------------
// Read head: content lookup on M_n (4 keys), mode mixing, read vectors.
// One block per batch. The batch's 128KB M_n tile is DMA'd into dynamic LDS
// by the Tensor Data Mover at kernel entry, overlapping the similarity/softmax
// phase (which streams the same tile through the cache path). The read_v pass
// then consumes the LDS copy with conflict-free bank access (w across lanes).
// ---------------------------------------------------------------------------
__global__ __launch_bounds__(512) void k_read_head(
    const float* __restrict__ if1, const float* __restrict__ Mn,
    const float* __restrict__ Wfwd, const float* __restrict__ Wbwd,
    float* __restrict__ Wr_out, float* __restrict__ readv)
{
    extern __shared__ float mnS[];        // dynamic LDS: Ndim*Wd floats (TDM dest)
    __shared__ float nkS[Rr][Wd];
    __shared__ float bR[Rr], mo[3][Rr];
    __shared__ float red[Ndim];
    __shared__ float wrL[Ndim * Rr];

    const int b = blockIdx.x, tid = threadIdx.x;
    const float* f = if1 + (size_t)b * IFACE;

    // Issue TDM: 2D tile 512x64 f32, contiguous, into dynamic-LDS base.
    if (tid == 0) {
        unsigned long long ga = (unsigned long long)(const void*)(Mn + (size_t)b * Ndim * Wd);
        u32x4 g0;
        g0[0] = 1u;                                   // count=1 (valid user D#)
        g0[1] = __builtin_amdgcn_groupstaticsize();   // lds_addr = dynamic LDS base
        g0[2] = (unsigned)ga;                         // global_addr[31:0]
        g0[3] = ((unsigned)(ga >> 32) & 0x01FFFFFFu) | (2u << 30); // addr[56:32] | type=2
        i32x8 g1;
        g1[0] = 2 << 16;                              // data_size = 4B
        g1[1] = Wd << 16;                             // tensor_dim0[15:0] = 64
        g1[2] = Ndim << 16;                           // dim0 hi=0 | tensor_dim1 lo = 512
        g1[3] = Wd << 16;                             // dim1 hi=0 | tile_dim0 = 64
        g1[4] = Ndim;                                 // tile_dim1 = 512, tile_dim2 = 0
        g1[5] = Wd;                                   // tensor_dim0_stride = 64
        g1[6] = (int)(((unsigned)(Ndim * Wd) & 0xFFFFu) << 16); // dim1_stride lo16
        g1[7] = 0;
        i32x4 z4 = {0, 0, 0, 0};
        i32x8 z8 = {0, 0, 0, 0, 0, 0, 0, 0};
        __builtin_amdgcn_tensor_load_to_lds(g0, g1, z4, z4, z8, 0);
    }

    if (tid < Rr) {
        float s = 0.0f;
        for (int w = 0; w < Wd; ++w) { float k = f[OFF_KREAD + tid*Wd + w]; s += k*k; }
        float inv = 1.0f / sqrtf(fmaxf(s, 1e-12f));
        for (int w = 0; w < Wd; ++w) nkS[tid][w] = f[OFF_KREAD + tid*Wd + w] * inv;
        bR[tid] = 1.0f + softplusf_(f[OFF_BREAD + tid]);
    }
    if (tid == 4) {  // modes softmax over axis of size 3
        for (int r = 0; r < Rr; ++r) {
            float m0 = f[OFF_MODES + 0*Rr + r], m1 = f[OFF_MODES + 1*Rr + r], m2 = f[OFF_MODES + 2*Rr + r];
            float mm = fmaxf(m0, fmaxf(m1, m2));
            float e0 = expf(m0-mm), e1 = expf(m1-mm), e2 = expf(m2-mm);
            float si = e0 + e1 + e2;
            mo[0][r] = e0/si; mo[1][r] = e1/si; mo[2][r] = e2/si;
        }
    }
    __syncthreads();

    const float* mrow = Mn + ((size_t)b * Ndim + tid) * Wd;
    float msq = 0.0f, dot[Rr] = {0.f, 0.f, 0.f, 0.f};
    for (int w = 0; w < Wd; ++w) {
        float mvv = mrow[w];
        msq += mvv * mvv;
#pragma unroll
        for (int r = 0; r < Rr; ++r) dot[r] += mvv * nkS[r][w];
    }
    float inv = 1.0f / sqrtf(fmaxf(msq, 1e-12f));
    float sim[Rr], wl[Rr];
#pragma unroll
    for (int r = 0; r < Rr; ++r) sim[r] = dot[r] * inv * bR[r];

    for (int r = 0; r < Rr; ++r) {
        red[tid] = sim[r]; __syncthreads();
        for (int s = Ndim/2; s > 0; s >>= 1) { if (tid < s) red[tid] = fmaxf(red[tid], red[tid+s]); __syncthreads(); }
        float mxv = red[0]; __syncthreads();
        float e = expf(sim[r] - mxv);
        red[tid] = e; __syncthreads();
        for (int s = Ndim/2; s > 0; s >>= 1) { if (tid < s) red[tid] += red[tid+s]; __syncthreads(); }
        wl[r] = e / red[0]; __syncthreads();
    }

#pragma unroll
    for (int r = 0; r < Rr; ++r) {
        float wrn = mo[0][r] * Wbwd[((size_t)b*Ndim + tid)*Rr + r]
                  + mo[1][r] * wl[r]
                  + mo[2][r] * Wfwd[((size_t)b*Ndim + tid)*Rr + r];
        Wr_out[((size_t)b*Ndim + tid)*Rr + r] = wrn;
        wrL[tid * Rr + r] = wrn;
    }

    // TDM data needed past this point: wave 0 drains TENSORcnt, barrier
    // publishes the LDS tile to all 16 waves.
    __builtin_amdgcn_s_wait_tensorcnt(0);
    __syncthreads();

    // read_v[r][w] = sum_n Mn[n][w] * Wr[n][r]   (Mn from LDS tile)
    if (tid < Rr * Wd) {
        int r = tid >> 6, w = tid & 63;
        float acc = 0.0f;
        for (int n = 0; n < Ndim; ++n)
            acc += mnS[n * Wd + w] * wrL[n * Rr + r];
        readv[(size_t)b * Rr * Wd + r * Wd + w] = acc;
    }
}

// ---------------------------------------------------------------------------
static inline int ceilDiv(int a, int b) { return (a + b - 1) / b; }

extern "C" void kernel_launch(void* const* d_in, const int* in_sizes, int n_in,
                              void* d_out, int out_size, void* d_ws, size_t ws_size,
                              hipStream_t stream) {
    (void)in_sizes; (void)n_in; (void)out_size; (void)ws_size;
    const float* inputs   = (const float*)d_in[0];
    const float* M        = (const float*)d_in[1];
    const float* usage    = (const float*)d_in[2];
    const float* L        = (const float*)d_in[3];
    const float* Wp       = (const float*)d_in[4];
    const float* Wread    = (const float*)d_in[5];
    const float* Wwrite   = (const float*)d_in[6];
    const float* hctrl    = (const float*)d_in[7];
    const float* Wiface   = (const float*)d_in[8];
    const float* Wreadout = (const float*)d_in[9];
    const float* gruK     = (const float*)d_in[10];
    const float* gruRK    = (const float*)d_in[11];
    const float* gruB     = (const float*)d_in[12];

    // outputs concatenated flat: y, M_n, usage_n, L_n, Wp_n, Wr_n, Ww, hc
    float* out    = (float*)d_out;
    float* y_out  = out;
    float* Mn_out = y_out  + (size_t)BSz * OUTD;
    float* un_out = Mn_out + (size_t)BSz * Ndim * Wd;
    float* Ln_out = un_out + (size_t)BSz * Ndim;
    float* Wp_out = Ln_out + (size_t)BSz * Ndim * Ndim;
    float* Wr_out = Wp_out + (size_t)BSz * Ndim;
    float* Ww_out = Wr_out + (size_t)BSz * Ndim * Rr;
    float* hc_out = Ww_out + (size_t)BSz * Ndim;

    // workspace
    float* ws   = (float*)d_ws;
    float* mx1  = ws;                          // BS*G3
    float* mh1  = mx1  + (size_t)BSz * G3;
    float* mx2  = mh1  + (size_t)BSz * G3;
    float* mh2  = mx2  + (size_t)BSz * G3;
    float* c1   = mh2  + (size_t)BSz * G3;     // BS*CDIM
    float* c2   = c1   + (size_t)BSz * CDIM;
    float* if1  = c2   + (size_t)BSz * CDIM;   // BS*IFACE
    float* if2  = if1  + (size_t)BSz * IFACE;
    float* Wwws = if2  + (size_t)BSz * IFACE;  // BS*N
    float* Wfwd = Wwws + (size_t)BSz * Ndim;   // BS*N*R
    float* Wbwd = Wfwd + (size_t)BSz * Ndim * Rr;
    float* rdv  = Wbwd + (size_t)BSz * Ndim * Rr; // BS*R*Wd

    dim3 w32(32);

    // GRU pre-activations (WMMA f32 GEMMs)
    {
        dim3 g(ceilDiv(G3, 16), ceilDiv(BSz, 16));
        k_gemm_wmma<<<g, w32, 0, stream>>>(inputs,        2*IND,  gruK,  G3, gruB,      mx1, G3, BSz, G3, IND);
        k_gemm_wmma<<<g, w32, 0, stream>>>(hctrl,         2*CDIM, gruRK, G3, gruB + G3, mh1, G3, BSz, G3, CDIM);
        k_gemm_wmma<<<g, w32, 0, stream>>>(inputs + IND,  2*IND,  gruK,  G3, gruB,      mx2, G3, BSz, G3, IND);
        k_gemm_wmma<<<g, w32, 0, stream>>>(hctrl + CDIM,  2*CDIM, gruRK, G3, gruB + G3, mh2, G3, BSz, G3, CDIM);
    }
    k_gru_combine<<<ceilDiv(BSz*CDIM, 256), 256, 0, stream>>>(mx1, mh1, hctrl, c1, hc_out, 0);
    k_gru_combine<<<ceilDiv(BSz*CDIM, 256), 256, 0, stream>>>(mx2, mh2, hctrl, c2, hc_out, 1);

    // interface projections
    {
        dim3 g(ceilDiv(IFACE, 16), ceilDiv(BSz, 16));
        k_gemm_wmma<<<g, w32, 0, stream>>>(c1, CDIM, Wiface, IFACE, nullptr, if1, IFACE, BSz, IFACE, CDIM);
        k_gemm_wmma<<<g, w32, 0, stream>>>(c2, CDIM, Wiface, IFACE, nullptr, if2, IFACE, BSz, IFACE, CDIM);
    }

    // write head (usage, allocator sort, write weighting, M_n, Wp_n)
    k_write_head<<<BSz, 512, 0, stream>>>(if2, M, usage, Wread, Wwrite, Wp,
                                          un_out, Ww_out, Wp_out, Mn_out, Wwws);

    // fused link-matrix update + forward/backward weightings
    k_zero<<<ceilDiv(BSz*Ndim*Rr, 256), 256, 0, stream>>>(Wbwd, BSz*Ndim*Rr);
    k_Ln<<<dim3(BSz, Ndim/32), 256, 0, stream>>>(L, Wwws, Wp, Wread, Ln_out, Wfwd, Wbwd);

    // read head (lookup on M_n, mode mixing, read vectors); dynamic LDS = M_n tile
    k_read_head<<<BSz, 512, Ndim * Wd * (int)sizeof(float), stream>>>(
        if1, Mn_out, Wfwd, Wbwd, Wr_out, rdv);

    // output projection
    {
        dim3 g(ceilDiv(OUTD, 16), ceilDiv(BSz, 16));
        k_gemm_wmma<<<g, w32, 0, stream>>>(rdv, Rr*Wd, Wreadout, OUTD, nullptr, y_out, OUTD, BSz, OUTD, Rr*Wd);
    }
}